// Matcher_87600152969549
// MI455X (gfx1250) — compile-verified
//
#include <hip/hip_runtime.h>
#include <hip/hip_bf16.h>
#include <math.h>
#include <stdint.h>

typedef float v2f __attribute__((ext_vector_type(2)));
typedef float v8f __attribute__((ext_vector_type(8)));

#define LC      4800            // coarse positions (60*80)
#define CF_OFF  23040000        // 4800*4800
#define ROWSX   51200           // 2048*25
#define INV_COARSE 0.0390625f   // 1/25.6

__device__ __forceinline__ v8f wmma4(v2f a, v2f b, v8f c) {
  // D = A(16x4 f32) * B(4x16 f32) + C(16x16 f32)
  return __builtin_amdgcn_wmma_f32_16x16x4_f32(false, a, false, b, (short)0, c,
                                               false, false);
}

__device__ __forceinline__ float geluf(float x) {
  float x3 = x * x * x;
  return 0.5f * x * (1.0f + tanhf(0.7978845608028654f * (x + 0.044715f * x3)));
}

// Async copy of 16 bytes from global to LDS (per-lane), tracked by ASYNCcnt.
__device__ __forceinline__ void async_b128(uint32_t lds_addr, uint64_t gaddr) {
  asm volatile("global_load_async_to_lds_b128 %0, %1, off"
               :: "v"(lds_addr), "v"(gaddr) : "memory");
}
__device__ __forceinline__ void wait_async0() {
  asm volatile("s_wait_asynccnt 0x0" ::: "memory");
}

// ---------------------------------------------------------------------------
// Workgroup-cooperative GEMM for N=K=128 (Q/K/V/O projections).
// Block = 256 threads (8 waves) computes a 16-row x 128-col strip.
// B (128x128, 64KB) staged via async-to-LDS b128; A tile via float4 loads.
// flags: 1 = +bias[col], 2 = +resid[row,col]
// ---------------------------------------------------------------------------
__global__ void gemm128_block(const float* __restrict__ A, const float* __restrict__ B,
                              const float* __restrict__ bias, const float* __restrict__ resid,
                              float* __restrict__ C, int flags) {
  __shared__ float as_[16 * 128];     // 8 KB
  __shared__ float bs_[128 * 128];    // 64 KB
  int tid = threadIdx.x, r0 = blockIdx.x * 16;

  // stage B asynchronously: 16384 floats = 4096 x b128
  uint32_t bs_lds = (uint32_t)(uintptr_t)bs_;
  uint64_t gB = (uint64_t)(uintptr_t)B;
  for (int i = tid; i < 4096; i += 256)
    async_b128(bs_lds + (uint32_t)i * 16u, gB + (uint64_t)i * 16u);

  // stage A tile with wide coalesced loads
  const float4* Ag = (const float4*)(A + (size_t)r0 * 128);
  float4* As4 = (float4*)as_;
  for (int i = tid; i < 512; i += 256) As4[i] = Ag[i];

  wait_async0();
  __syncthreads();

  int wave = tid >> 5, lane = tid & 31, half = lane >> 4, lm = lane & 15;
  const float* Arow = as_ + lm * 128;
  const float* Bcol = bs_ + wave * 16 + lm;
  v8f acc = {};
#pragma unroll 4
  for (int k0 = 0; k0 < 128; k0 += 4) {
    int ka = k0 + 2 * half;
    v2f a = { Arow[ka], Arow[ka + 1] };            // ds_load_b64
    v2f b = { Bcol[ka * 128], Bcol[(ka + 1) * 128] };
    acc = wmma4(a, b, acc);
  }
  int col = wave * 16 + lm;
  float bc = (flags & 1) ? bias[col] : 0.0f;
#pragma unroll
  for (int r = 0; r < 8; ++r) {
    int row = r0 + r + 8 * half;
    float v = acc[r] + bc;
    if (flags & 2) v += resid[(size_t)row * 128 + col];
    C[(size_t)row * 128 + col] = v;
  }
}

// ---------------------------------------------------------------------------
// fcw = [c0[i_ids]; c1[j_ids]] (2048x256) @ Wdown (256x128) + bdown
// ---------------------------------------------------------------------------
__global__ void fcw_gemm(const float* __restrict__ feat_c, const int* __restrict__ i_ids,
                         const int* __restrict__ j_ids, const float* __restrict__ Wdown,
                         const float* __restrict__ bdown, float* __restrict__ fcw) {
  int wave = (blockIdx.x * blockDim.x + threadIdx.x) >> 5;
  if (wave >= 128 * 8) return;
  int lane = threadIdx.x & 31, half = lane >> 4, lm = lane & 15;
  int tr = wave / 8, tc = wave % 8;
  int row = tr * 16 + lm;
  int id = (row < 1024) ? i_ids[row] : (LC + j_ids[row - 1024]);
  const float* Arow = feat_c + (size_t)id * 256;
  const float* Bcol = Wdown + (tc * 16 + lm);
  v8f acc = {};
  for (int k0 = 0; k0 < 256; k0 += 4) {
    int ka = k0 + 2 * half;
    v2f a = { Arow[ka], Arow[ka + 1] };
    v2f b = { Bcol[(size_t)ka * 128], Bcol[(size_t)(ka + 1) * 128] };
    acc = wmma4(a, b, acc);
  }
  int col = tc * 16 + lm;
  float bc = bdown[col];
#pragma unroll
  for (int r = 0; r < 8; ++r) {
    int orow = tr * 16 + r + 8 * half;
    fcw[(size_t)orow * 128 + col] = acc[r] + bc;
  }
}

// ---------------------------------------------------------------------------
// X = [patch_gather(feat_f) | broadcast fcw] (51200x256) @ Wmerge + bmerge
// ---------------------------------------------------------------------------
__global__ void merge_gemm(const float* __restrict__ feat_f, const float* __restrict__ fcw,
                           const int* __restrict__ i_ids, const int* __restrict__ j_ids,
                           const float* __restrict__ Wmerge, const float* __restrict__ bmerge,
                           float* __restrict__ X) {
  int wave = (blockIdx.x * blockDim.x + threadIdx.x) >> 5;
  if (wave >= 3200 * 8) return;
  int lane = threadIdx.x & 31, half = lane >> 4, lm = lane & 15;
  int tr = wave / 8, tc = wave % 8;
  int row = tr * 16 + lm;
  int mp = row / 25, w = row % 25;
  int id, img;
  if (mp < 1024) { id = i_ids[mp]; img = 0; } else { id = j_ids[mp - 1024]; img = 1; }
  int py = id / 80, px = id % 80, wy = w / 5, wx = w % 5;
  int y = py * 4 - 2 + wy, x = px * 4 - 2 + wx;
  bool inb = (y >= 0 && y < 240 && x >= 0 && x < 320);
  const float* pf = feat_f + (size_t)img * 9830400 + (size_t)(inb ? (y * 320 + x) : 0);
  const float* pw = fcw + (size_t)mp * 128;
  const float* Bcol = Wmerge + (tc * 16 + lm);
  v8f acc = {};
  for (int k0 = 0; k0 < 256; k0 += 4) {
    int ka = k0 + 2 * half;
    float a0 = (ka     < 128) ? (inb ? pf[(size_t)ka * 76800] : 0.0f) : pw[ka - 128];
    float a1 = (ka + 1 < 128) ? (inb ? pf[(size_t)(ka + 1) * 76800] : 0.0f) : pw[ka + 1 - 128];
    v2f a = { a0, a1 };
    v2f b = { Bcol[(size_t)ka * 128], Bcol[(size_t)(ka + 1) * 128] };
    acc = wmma4(a, b, acc);
  }
  int col = tc * 16 + lm;
  float bc = bmerge[col];
#pragma unroll
  for (int r = 0; r < 8; ++r) {
    int orow = tr * 16 + r + 8 * half;
    X[(size_t)orow * 128 + col] = acc[r] + bc;
  }
}

// ---------------------------------------------------------------------------
// Cross-attention core: one block per (match, direction); S=25, NH=4, dh=32
// ---------------------------------------------------------------------------
__global__ void attn_core(const float* __restrict__ Q, const float* __restrict__ Kb,
                          const float* __restrict__ V, float* __restrict__ Attn) {
  __shared__ float Qs[25 * 128], Ks[25 * 128], Vs[25 * 128];
  int m2 = blockIdx.x, tid = threadIdx.x;
  size_t qbase = (size_t)m2 * 25 * 128;
  size_t kvbase = (size_t)((m2 < 1024) ? (1024 + m2) : (m2 - 1024)) * 25 * 128;
  for (int i = tid; i < 3200; i += 128) {
    Qs[i] = Q[qbase + i]; Ks[i] = Kb[kvbase + i]; Vs[i] = V[kvbase + i];
  }
  __syncthreads();
  if (tid < 100) {
    int h = tid / 25, qi = tid % 25;
    const float* qp = Qs + qi * 128 + h * 32;
    float s[25], mx = -1e30f;
#pragma unroll 1
    for (int r = 0; r < 25; ++r) {
      const float* kp = Ks + r * 128 + h * 32;
      float acc = 0.0f;
#pragma unroll
      for (int d = 0; d < 32; ++d) acc += qp[d] * kp[d];
      s[r] = acc * 0.17677669529663687f;      // 1/sqrt(32)
      mx = fmaxf(mx, s[r]);
    }
    float sum = 0.0f;
#pragma unroll
    for (int r = 0; r < 25; ++r) { s[r] = __expf(s[r] - mx); sum += s[r]; }
    float inv = 1.0f / sum;
#pragma unroll 1
    for (int d = 0; d < 32; ++d) {
      float o = 0.0f;
#pragma unroll
      for (int r = 0; r < 25; ++r) o += s[r] * Vs[r * 128 + h * 32 + d];
      Attn[qbase + (size_t)qi * 128 + h * 32 + d] = o * inv;
    }
  }
}

// ---------------------------------------------------------------------------
// LayerNorm over C=128; one wave per row
// ---------------------------------------------------------------------------
__global__ void ln_kernel(const float* __restrict__ in, const float* __restrict__ g,
                          const float* __restrict__ b, float* __restrict__ out, int rows) {
  int wv = (blockIdx.x * blockDim.x + threadIdx.x) >> 5;
  if (wv >= rows) return;
  int lane = threadIdx.x & 31, c = lane * 4;
  float4 v = *(const float4*)(in + (size_t)wv * 128 + c);
  float s = v.x + v.y + v.z + v.w;
  for (int msk = 1; msk < 32; msk <<= 1) s += __shfl_xor(s, msk, 32);
  float mean = s * (1.0f / 128.0f);
  float d0 = v.x - mean, d1 = v.y - mean, d2 = v.z - mean, d3 = v.w - mean;
  float q = d0 * d0 + d1 * d1 + d2 * d2 + d3 * d3;
  for (int msk = 1; msk < 32; msk <<= 1) q += __shfl_xor(q, msk, 32);
  float rs = rsqrtf(q * (1.0f / 128.0f) + 1e-5f);
  float4 gg = *(const float4*)(g + c);
  float4 bb = *(const float4*)(b + c);
  float4 o = { d0 * rs * gg.x + bb.x, d1 * rs * gg.y + bb.y,
               d2 * rs * gg.z + bb.z, d3 * rs * gg.w + bb.w };
  *(float4*)(out + (size_t)wv * 128 + c) = o;
}

// ---------------------------------------------------------------------------
// Fused MLP: out = Xa + gelu(Xa@W1 + b1)@W2 + b2 ; 16 rows per block (8 waves)
// Xa tile staged via async-to-LDS.
// ---------------------------------------------------------------------------
__global__ void mlp_fused(const float* __restrict__ Xa, const float* __restrict__ W1,
                          const float* __restrict__ b1, const float* __restrict__ W2,
                          const float* __restrict__ b2, float* __restrict__ out) {
  __shared__ float xs[16 * 128];
  __shared__ float hs[16 * 512];
  int tid = threadIdx.x, r0 = blockIdx.x * 16;
  uint32_t xs_lds = (uint32_t)(uintptr_t)xs;
  uint64_t gX = (uint64_t)(uintptr_t)(Xa + (size_t)r0 * 128);
  for (int i = tid; i < 512; i += 256)
    async_b128(xs_lds + (uint32_t)i * 16u, gX + (uint64_t)i * 16u);
  wait_async0();
  __syncthreads();
  int wave = tid >> 5, lane = tid & 31, half = lane >> 4, lm = lane & 15;
  // stage 1: h = gelu(xs @ W1 + b1); each wave does cols [wave*64, wave*64+64)
  for (int t = 0; t < 4; ++t) {
    int cb = wave * 64 + t * 16;
    const float* Arow = xs + lm * 128;
    const float* Bcol = W1 + cb + lm;
    v8f acc = {};
    for (int k0 = 0; k0 < 128; k0 += 4) {
      int ka = k0 + 2 * half;
      v2f a = { Arow[ka], Arow[ka + 1] };
      v2f b = { Bcol[(size_t)ka * 512], Bcol[(size_t)(ka + 1) * 512] };
      acc = wmma4(a, b, acc);
    }
    int col = cb + lm;
    float bc = b1[col];
#pragma unroll
    for (int r = 0; r < 8; ++r) hs[(r + 8 * half) * 512 + col] = geluf(acc[r] + bc);
  }
  __syncthreads();
  // stage 2: out = hs @ W2 + b2 + xs; each wave does cols [wave*16, wave*16+16)
  {
    int cb = wave * 16;
    const float* Arow = hs + lm * 512;
    const float* Bcol = W2 + cb + lm;
    v8f acc = {};
    for (int k0 = 0; k0 < 512; k0 += 4) {
      int ka = k0 + 2 * half;
      v2f a = { Arow[ka], Arow[ka + 1] };
      v2f b = { Bcol[(size_t)ka * 128], Bcol[(size_t)(ka + 1) * 128] };
      acc = wmma4(a, b, acc);
    }
    int col = cb + lm;
    float bc = b2[col];
#pragma unroll
    for (int r = 0; r < 8; ++r) {
      int rr = r + 8 * half;
      out[(size_t)(r0 + rr) * 128 + col] = acc[r] + bc + xs[rr * 128 + col];
    }
  }
}

// ---------------------------------------------------------------------------
// Final 5x5 soft-argmax -> rcoords
// ---------------------------------------------------------------------------
__global__ void coords_kernel(const float* __restrict__ Y, const int* __restrict__ j_ids,
                              float* __restrict__ rc) {
  int m = blockIdx.x * blockDim.x + threadIdx.x;
  if (m >= 1024) return;
  const float* f0 = Y + ((size_t)m * 25 + 12) * 128;
  const float* y1 = Y + (size_t)(1024 + m) * 25 * 128;
  float s[25], mx = -1e30f;
#pragma unroll 1
  for (int r = 0; r < 25; ++r) {
    const float* yr = y1 + r * 128;
    float acc = 0.0f;
    for (int c = 0; c < 128; ++c) acc += f0[c] * yr[c];
    s[r] = acc * 0.08838834764831845f;        // 1/sqrt(128)
    mx = fmaxf(mx, s[r]);
  }
  float sum = 0.0f;
#pragma unroll
  for (int r = 0; r < 25; ++r) { s[r] = __expf(s[r] - mx); sum += s[r]; }
  float inv = 1.0f / sum, cx = 0.0f, cy = 0.0f;
#pragma unroll
  for (int r = 0; r < 25; ++r) {
    float h = s[r] * inv;
    cx += h * (-1.0f + 0.5f * (float)(r % 5));
    cy += h * (-1.0f + 0.5f * (float)(r / 5));
  }
  int j = j_ids[m];
  rc[(size_t)m * 2 + 0] = ((float)(j % 80) * 4.0f + cx * 2.0f) * 2.0f;
  rc[(size_t)m * 2 + 1] = ((float)(j / 80) * 4.0f + cy * 2.0f) * 2.0f;
}

// ---------------------------------------------------------------------------
// Online softmax stats of rows of A@B^T * INV_COARSE (A,B: 4800x256)
// Wave owns 16 rows, sweeps all 300 column tiles.
// ---------------------------------------------------------------------------
__global__ void sim_stats(const float* __restrict__ A, const float* __restrict__ B,
                          float* __restrict__ omax, float* __restrict__ osum) {
  int wave = (blockIdx.x * blockDim.x + threadIdx.x) >> 5;
  if (wave >= 300) return;
  int lane = threadIdx.x & 31, half = lane >> 4, lm = lane & 15;
  int r0 = wave * 16;
  const float* Arow = A + (size_t)(r0 + lm) * 256;
  float rmax[8], rsum[8];
#pragma unroll
  for (int r = 0; r < 8; ++r) { rmax[r] = -1e30f; rsum[r] = 0.0f; }
  for (int s0 = 0; s0 < LC; s0 += 16) {
    const float* Brow = B + (size_t)(s0 + lm) * 256;   // B^T access: lane lm = col
    v8f acc = {};
    for (int k0 = 0; k0 < 256; k0 += 4) {
      int ka = k0 + 2 * half;
      v2f a = { Arow[ka], Arow[ka + 1] };
      v2f b = { Brow[ka], Brow[ka + 1] };
      acc = wmma4(a, b, acc);
    }
#pragma unroll
    for (int r = 0; r < 8; ++r) {
      float t = acc[r] * INV_COARSE;
      float tm = t;
      for (int msk = 1; msk < 16; msk <<= 1) tm = fmaxf(tm, __shfl_xor(tm, msk, 32));
      float nm = fmaxf(rmax[r], tm);
      float e = __expf(t - nm);
      for (int msk = 1; msk < 16; msk <<= 1) e += __shfl_xor(e, msk, 32);
      rsum[r] = rsum[r] * __expf(rmax[r] - nm) + e;
      rmax[r] = nm;
    }
  }
  if (lm == 0) {
#pragma unroll
    for (int r = 0; r < 8; ++r) {
      int row = r0 + r + 8 * half;
      omax[row] = rmax[r]; osum[row] = rsum[r];
    }
  }
}

// ---------------------------------------------------------------------------
// cf tile: recompute sm tile, apply dual softmax, write to d_out
// ---------------------------------------------------------------------------
__global__ void cf_kernel(const float* __restrict__ A, const float* __restrict__ B,
                          const float* __restrict__ rowmax, const float* __restrict__ rowsum,
                          const float* __restrict__ colmax, const float* __restrict__ colsum,
                          float* __restrict__ cf) {
  int wave = (blockIdx.x * blockDim.x + threadIdx.x) >> 5;
  if (wave >= 300 * 300) return;
  int lane = threadIdx.x & 31, half = lane >> 4, lm = lane & 15;
  int tr = wave / 300, tc = wave % 300;
  const float* Arow = A + (size_t)(tr * 16 + lm) * 256;
  const float* Brow = B + (size_t)(tc * 16 + lm) * 256;
  v8f acc = {};
  for (int k0 = 0; k0 < 256; k0 += 4) {
    int ka = k0 + 2 * half;
    v2f a = { Arow[ka], Arow[ka + 1] };
    v2f b = { Brow[ka], Brow[ka + 1] };
    acc = wmma4(a, b, acc);
  }
  int col = tc * 16 + lm;
  float cm = colmax[col], ci = 1.0f / colsum[col];
#pragma unroll
  for (int r = 0; r < 8; ++r) {
    int row = tr * 16 + r + 8 * half;
    float s = acc[r] * INV_COARSE;
    float v = __expf(s - rowmax[row]) / rowsum[row] * __expf(s - cm) * ci;
    cf[(size_t)row * LC + col] = v;
  }
}

// ---------------------------------------------------------------------------
extern "C" void kernel_launch(void* const* d_in, const int* in_sizes, int n_in,
                              void* d_out, int out_size, void* d_ws, size_t ws_size,
                              hipStream_t stream) {
  (void)in_sizes; (void)n_in; (void)out_size; (void)ws_size;
  const float* feat_f = (const float*)d_in[0];
  const float* feat_c = (const float*)d_in[1];
  const int*   i_ids  = (const int*)d_in[3];     // b_ids (d_in[2]) are all 0
  const int*   j_ids  = (const int*)d_in[4];
  const float* Wdown  = (const float*)d_in[5];
  const float* bdown  = (const float*)d_in[6];
  const float* Wmerge = (const float*)d_in[7];
  const float* bmerge = (const float*)d_in[8];
  const float* Wq     = (const float*)d_in[9];
  const float* Wk     = (const float*)d_in[10];
  const float* Wv     = (const float*)d_in[11];
  const float* Wo     = (const float*)d_in[12];
  const float* W1     = (const float*)d_in[13];
  const float* b1     = (const float*)d_in[14];
  const float* W2     = (const float*)d_in[15];
  const float* b2     = (const float*)d_in[16];
  const float* ln1g   = (const float*)d_in[17];
  const float* ln1b   = (const float*)d_in[18];
  const float* ln2g   = (const float*)d_in[19];
  const float* ln2b   = (const float*)d_in[20];

  float* out = (float*)d_out;
  // big intermediates live inside the (not-yet-written) cf region of d_out
  float* X  = out;               // 6,553,600 floats
  float* Qb = out + 6553600;     // 6,553,600
  float* Kb = out + 13107200;    // 6,553,600  (ends < CF_OFF)

  float* ws = (float*)d_ws;
  float* rowmax = ws;
  float* rowsum = ws + 4800;
  float* colmax = ws + 9600;
  float* colsum = ws + 14400;
  float* fcw    = ws + 19200;                 // 2048*128
  float* Vbuf   = ws + 19200 + 262144;        // 51200*128
  float* Attn   = Vbuf + 6553600;             // 51200*128

  const float* c0 = feat_c;
  const float* c1 = feat_c + (size_t)LC * 256;

  // ---- fine pipeline (uses d_out as scratch; must precede cf write) ----
  fcw_gemm  <<<128, 256, 0, stream>>>(feat_c, i_ids, j_ids, Wdown, bdown, fcw);
  merge_gemm<<<3200, 256, 0, stream>>>(feat_f, fcw, i_ids, j_ids, Wmerge, bmerge, X);
  gemm128_block<<<3200, 256, 0, stream>>>(X, Wq, nullptr, nullptr, Qb, 0);
  gemm128_block<<<3200, 256, 0, stream>>>(X, Wk, nullptr, nullptr, Kb, 0);
  gemm128_block<<<3200, 256, 0, stream>>>(X, Wv, nullptr, nullptr, Vbuf, 0);
  attn_core <<<2048, 128, 0, stream>>>(Qb, Kb, Vbuf, Attn);
  gemm128_block<<<3200, 256, 0, stream>>>(Attn, Wo, nullptr, X, Qb, 2);
  ln_kernel <<<6400, 256, 0, stream>>>(Qb, ln1g, ln1b, X, ROWSX);          // Xa -> X
  mlp_fused <<<3200, 256, 0, stream>>>(X, W1, b1, W2, b2, Qb);             // pre-LN2 -> Qb
  ln_kernel <<<6400, 256, 0, stream>>>(Qb, ln2g, ln2b, Kb, ROWSX);         // y -> Kb
  coords_kernel<<<4, 256, 0, stream>>>(Kb, j_ids, out + CF_OFF);

  // ---- coarse dual-softmax (overwrites scratch region with cf) ----
  sim_stats<<<75, 128, 0, stream>>>(c0, c1, rowmax, rowsum);  // axis=2 stats
  sim_stats<<<75, 128, 0, stream>>>(c1, c0, colmax, colsum);  // axis=1 stats
  cf_kernel<<<11250, 256, 0, stream>>>(c0, c1, rowmax, rowsum, colmax, colsum, out);
}